// SOMLayer_8864812499640
// MI455X (gfx1250) — compile-verified
//
#include <hip/hip_runtime.h>
#include <hip/hip_bf16.h>

// ---------------------------------------------------------------------------
// SOM layer for MI455X (gfx1250).  Dominant op: 32768x1024x256 f32 GEMM for
// cdist via V_WMMA_F32_16X16X4_F32.  WG = 128 rows x 1024 nodes: A tile lives
// in VGPRs (wave-private, 128 VGPR/lane), B tiles double-buffered in LDS and
// shared by all 8 waves (8x less L2 traffic).  q round-trips through L2.
// ---------------------------------------------------------------------------

typedef float v2f __attribute__((ext_vector_type(2)));
typedef float v8f __attribute__((ext_vector_type(8)));

#define LATENT   256
#define NNODES   1024
#define ROWS     32768          // 64 * 512
#define SEQ      512
#define APAD     260            // 256 + 4: bank-conflict pad, 16B-aligned rows
#define LOG2_DECAY (-0.15200309344504997f)   // log2(0.9)

// ---- output layout (floats) -----------------------------------------------
#define OUT_SOMZ   0
#define OUT_SCAL   8388608      // total, kl, div, tsmooth, neighbor
#define OUT_Q      8388613
#define OUT_BMU    (8388613 + 33554432)

// ---- workspace layout (floats) --------------------------------------------
// [0..1023]      colsum of q^2
// [1024..2047]   node norms
// [2048..2051]   accumulators: kl, div, tsmooth, neighbor
// [2052.. ]      bmu as int32 (32768)

__device__ inline void block_reduce_add(float v, float* target) {
    __shared__ float red[8];
    int lane = threadIdx.x & 31, wave = threadIdx.x >> 5;
#pragma unroll
    for (int off = 16; off; off >>= 1) v += __shfl_xor(v, off, 32);
    if (lane == 0) red[wave] = v;
    __syncthreads();
    if (threadIdx.x == 0) {
        float s = 0.f;
        int nw = blockDim.x >> 5;
        for (int w = 0; w < nw; ++w) s += red[w];
        atomicAdd(target, s);
    }
}

// ---------------------------------------------------------------------------
// init: node norms + zero accumulators (re-zeroed every launch: graph-safe)
// ---------------------------------------------------------------------------
__global__ __launch_bounds__(256) void som_init(const float* __restrict__ nodes,
                                                float* __restrict__ colsum,
                                                float* __restrict__ nodenorm,
                                                float* __restrict__ scal) {
    int i = blockIdx.x * 256 + threadIdx.x;      // 4 blocks x 256 = 1024
    const float* p = nodes + (long)i * LATENT;
    float s = 0.f;
#pragma unroll 8
    for (int d = 0; d < LATENT; ++d) s += p[d] * p[d];
    nodenorm[i] = s;
    colsum[i] = 0.f;
    if (i < 4) scal[i] = 0.f;
}

// ---------------------------------------------------------------------------
// phase 1: 256 WGs x 256 threads.  WG = 128 rows; wave w owns rows
// [wg*128 + w*16, +16) and sweeps all 64 node tiles.  B shared via LDS.
// ---------------------------------------------------------------------------
__global__ __launch_bounds__(256) void som_phase1(const float* __restrict__ z,
                                                  const float* __restrict__ nodes,
                                                  const float* __restrict__ nodenorm,
                                                  float* __restrict__ colsum,
                                                  float* __restrict__ som_z,
                                                  float* __restrict__ q_out,
                                                  float* __restrict__ bmu_f,
                                                  int* __restrict__ bmu_i) {
    __shared__ __align__(16) float Bs[2][16 * APAD];
    __shared__ float colsum_lds[NNODES];

    const int tid   = threadIdx.x;
    const int wave  = tid >> 5;
    const int lane  = tid & 31;
    const int lhalf = lane >> 4;      // k-pair half for A/B, row half for C/D
    const int l16   = lane & 15;
    const long gi0  = (long)blockIdx.x * 128 + wave * 16;   // wave's first row

#pragma unroll
    for (int c = 0; c < 4; ++c) colsum_lds[tid + 256 * c] = 0.f;

    // ---- A tile in registers: lane carries row l16, k-half lhalf ----------
    {
    }
    const long giA = gi0 + l16;
    const int  tt  = (int)(giA & (SEQ - 1));
    const float w  = exp2f((float)(SEQ - 1 - tt) * LOG2_DECAY);   // 0.9^(511-t)
    const float* zrow = z + giA * LATENT + 2 * lhalf;
    v2f av[64];
    float rn_part = 0.f;
#pragma unroll
    for (int j = 0; j < 64; ++j) {
        v2f a = *(const v2f*)(zrow + 4 * j);
        a.x *= w; a.y *= w;
        av[j] = a;
        rn_part += a.x * a.x + a.y * a.y;
    }
    const float rn_mine = rn_part + __shfl_xor(rn_part, 16, 32); // ||row l16||^2
    float rnorm[8];
#pragma unroll
    for (int r = 0; r < 8; ++r) rnorm[r] = __shfl(rn_mine, 8 * lhalf + r, 32);

    // ---- B staging: 16 nodes x 256, float4-coalesced ----------------------
    auto stageB = [&](int buf, int ct) {
        const int n0 = ct * 16;
        const int e0 = tid >> 6;          // 0..3
        const int c4 = (tid & 63) * 4;
#pragma unroll
        for (int p = 0; p < 4; ++p) {
            int e = e0 + p * 4;
            float4 v = *(const float4*)(nodes + (long)(n0 + e) * LATENT + c4);
            *(float4*)(&Bs[buf][e * APAD + c4]) = v;
        }
    };

    stageB(0, 0);
    __syncthreads();

    float rs[8], mv[8];
    int   mi[8];
#pragma unroll
    for (int r = 0; r < 8; ++r) { rs[r] = 0.f; mv[r] = 3.4e38f; mi[r] = 0; }

#pragma unroll 1
    for (int ct = 0; ct < 64; ++ct) {
        const int buf = ct & 1;
        if (ct < 63) stageB(buf ^ 1, ct + 1);

        const float* Bp = &Bs[buf][l16 * APAD + 2 * lhalf];
        v8f acc = {0.f, 0.f, 0.f, 0.f, 0.f, 0.f, 0.f, 0.f};
#pragma unroll
        for (int j = 0; j < 64; ++j) {
            v2f b = *(const v2f*)(Bp + 4 * j);
            acc = __builtin_amdgcn_wmma_f32_16x16x4_f32(
                false, av[j], false, b, (short)0, acc, false, false);
        }

        const int n = ct * 16 + l16;
        const float nn = nodenorm[n];
#pragma unroll
        for (int r = 0; r < 8; ++r) {
            float d2 = fmaxf(rnorm[r] + nn - 2.0f * acc[r], 0.f);
            float dist = (d2 > 0.f) ? sqrtf(d2) : 0.f;
            if (dist < mv[r]) { mv[r] = dist; mi[r] = n; }
            float qu = 1.0f / (1.0f + dist);        // ALPHA=1 -> exponent 1
            rs[r] += qu;
            long gi = gi0 + 8 * lhalf + r;
            q_out[gi * NNODES + n] = qu;            // unnormalized; fixed below
        }
        __syncthreads();                            // B double-buffer fence
    }

    // ---- wave-local row reductions (wave owns complete rows) --------------
    float inv_rs[8];
#pragma unroll
    for (int r = 0; r < 8; ++r) {
        float s = rs[r];
#pragma unroll
        for (int off = 1; off < 16; off <<= 1) s += __shfl_xor(s, off, 32);
        inv_rs[r] = 1.0f / s;
        float v = mv[r]; int i = mi[r];
#pragma unroll
        for (int off = 1; off < 16; off <<= 1) {
            float ov = __shfl_xor(v, off, 32);
            int   oi = __shfl_xor(i, off, 32);
            if (ov < v || (ov == v && oi < i)) { v = ov; i = oi; }
        }
        mv[r] = v; mi[r] = i;
        if (l16 == 0) {
            long gi = gi0 + 8 * lhalf + r;
            bmu_i[gi] = i;
            bmu_f[gi] = (float)i;
        }
    }

    // ---- normalize q in place (L2-hot), accumulate colsum(q^2) ------------
    float4 cs4[8];
#pragma unroll
    for (int c = 0; c < 8; ++c) cs4[c] = make_float4(0.f, 0.f, 0.f, 0.f);

#pragma unroll
    for (int e = 0; e < 16; ++e) {
        float inv = __shfl(inv_rs[e & 7], 16 * (e >> 3), 32);
        float* qrow = q_out + (gi0 + e) * NNODES;
#pragma unroll
        for (int c = 0; c < 8; ++c) {
            int col = lane * 4 + c * 128;
            float4 v = *(const float4*)(qrow + col);
            v.x *= inv; v.y *= inv; v.z *= inv; v.w *= inv;
            *(float4*)(qrow + col) = v;
            cs4[c].x += v.x * v.x; cs4[c].y += v.y * v.y;
            cs4[c].z += v.z * v.z; cs4[c].w += v.w * v.w;
        }
    }
#pragma unroll
    for (int c = 0; c < 8; ++c) {
        int col = lane * 4 + c * 128;
        atomicAdd(&colsum_lds[col + 0], cs4[c].x);
        atomicAdd(&colsum_lds[col + 1], cs4[c].y);
        atomicAdd(&colsum_lds[col + 2], cs4[c].z);
        atomicAdd(&colsum_lds[col + 3], cs4[c].w);
    }

    // ---- som_z = 0.9*z + 0.1*nodes[bmu] -----------------------------------
#pragma unroll
    for (int e = 0; e < 16; ++e) {
        int bmu_e = __shfl(mi[e & 7], 16 * (e >> 3), 32);
        long gi = gi0 + e;
        const float* zr = z + gi * LATENT;
        const float* nr = nodes + (long)bmu_e * LATENT;
        float* sr = som_z + gi * LATENT;
#pragma unroll
        for (int c = 0; c < 2; ++c) {
            int col = lane * 4 + c * 128;
            float4 zv = *(const float4*)(zr + col);
            float4 nv = *(const float4*)(nr + col);
            float4 o;
            o.x = 0.9f * zv.x + 0.1f * nv.x;
            o.y = 0.9f * zv.y + 0.1f * nv.y;
            o.z = 0.9f * zv.z + 0.1f * nv.z;
            o.w = 0.9f * zv.w + 0.1f * nv.w;
            *(float4*)(sr + col) = o;
        }
    }

    __syncthreads();                                 // colsum_lds complete
#pragma unroll
    for (int c = 0; c < 4; ++c)
        atomicAdd(&colsum[tid + 256 * c], colsum_lds[tid + 256 * c]);
}

// ---------------------------------------------------------------------------
// KL: one wave per 2 rows; r = q^2/cs_j, R = sum r, term = p(log p - log q)
//   = (r/R) * (0.5*log r - 0.5*log cs_j - log R)
// ---------------------------------------------------------------------------
__global__ __launch_bounds__(256) void som_kl(const float* __restrict__ q,
                                              const float* __restrict__ colsum,
                                              float* __restrict__ kl_acc) {
    const int lane = threadIdx.x & 31, wave = threadIdx.x >> 5;
    float wkl = 0.f;
#pragma unroll 1
    for (int rr = 0; rr < 2; ++rr) {
        long gi = (long)blockIdx.x * 16 + wave * 2 + rr;
        const float* qrow = q + gi * NNODES;
        float rv[32];
        float R = 0.f;
#pragma unroll
        for (int e = 0; e < 32; ++e) {
            int j = lane + e * 32;
            float qq = qrow[j];
            float r = qq * qq / colsum[j];
            rv[e] = r;
            R += r;
        }
#pragma unroll
        for (int off = 16; off; off >>= 1) R += __shfl_xor(R, off, 32);
        float logR = __logf(R), invR = 1.0f / R;
        float acc = 0.f;
#pragma unroll
        for (int e = 0; e < 32; ++e) {
            int j = lane + e * 32;
            float r = rv[e];
            acc += (r * invR) * (0.5f * __logf(r) - 0.5f * __logf(colsum[j]) - logR);
        }
        wkl += acc;
    }
    block_reduce_add(wkl, kl_acc);
}

// ---------------------------------------------------------------------------
__global__ __launch_bounds__(256) void som_tsmooth(const float* __restrict__ z,
                                                   float* __restrict__ ts_acc) {
    const long N = 64L * 511 * 256;
    long stride = (long)gridDim.x * blockDim.x;
    float acc = 0.f;
    for (long i = (long)blockIdx.x * blockDim.x + threadIdx.x; i < N; i += stride) {
        long b = i / (511 * 256);
        long rem = i - b * (511 * 256);        // t*256 + d, t in [0,511)
        long base = b * (long)(SEQ * LATENT) + rem;
        float d = z[base + LATENT] - z[base];
        acc += d * d;
    }
    block_reduce_add(acc, ts_acc);
}

// ---------------------------------------------------------------------------
// diversity: -mean(cdist(nodes, nodes)).  64 WGs, 16 i-rows each, j LDS-blocked.
// ---------------------------------------------------------------------------
__global__ __launch_bounds__(256) void som_div(const float* __restrict__ nodes,
                                               const float* __restrict__ nodenorm,
                                               float* __restrict__ div_acc) {
    __shared__ float ilds[16][LATENT + 1];
    __shared__ float jlds[64][LATENT + 1];
    const int tid = threadIdx.x;
    const int i0 = blockIdx.x * 16;
    const int ti = tid & 15;          // i within block
    for (int e = 0; e < 16; ++e)
        ilds[e][tid] = nodes[(long)(i0 + e) * LATENT + tid];
    const float ni = nodenorm[i0 + ti];
    float acc = 0.f;
#pragma unroll 1
    for (int j0 = 0; j0 < NNODES; j0 += 64) {
        __syncthreads();
        for (int e = 0; e < 64; ++e)
            jlds[e][tid] = nodes[(long)(j0 + e) * LATENT + tid];
        __syncthreads();
#pragma unroll 1
        for (int pass = 0; pass < 4; ++pass) {
            int jj = (tid >> 4) + pass * 16;
            float dot = 0.f;
#pragma unroll 8
            for (int d = 0; d < LATENT; ++d) dot += ilds[ti][d] * jlds[jj][d];
            float d2 = fmaxf(ni + nodenorm[j0 + jj] - 2.0f * dot, 0.f);
            acc += (d2 > 0.f) ? sqrtf(d2) : 0.f;
        }
    }
    __syncthreads();
    block_reduce_add(acc, div_acc);
}

// ---------------------------------------------------------------------------
__global__ __launch_bounds__(256) void som_neigh(const int* __restrict__ bmu,
                                                 float* __restrict__ nb_acc) {
    const int NP = 64 * 511;
    int stride = gridDim.x * blockDim.x;
    float acc = 0.f;
    for (int i = blockIdx.x * blockDim.x + threadIdx.x; i < NP; i += stride) {
        int b = i / 511, t = i - b * 511;
        int p = bmu[b * SEQ + t], n = bmu[b * SEQ + t + 1];
        acc += (float)(abs((p >> 5) - (n >> 5)) + abs((p & 31) - (n & 31)));
    }
    block_reduce_add(acc, nb_acc);
}

// ---------------------------------------------------------------------------
__global__ void som_final(const float* __restrict__ scal, float* __restrict__ out) {
    float kl = scal[0] / 32768.0f;
    float dv = -scal[1] / (1024.0f * 1024.0f);
    float ts = scal[2] / (64.0f * 511.0f * 256.0f) * 0.9f;
    float nb = scal[3] / (64.0f * 511.0f);
    out[0] = kl + 0.5f * dv + 0.3f * ts + 0.2f * nb;
    out[1] = kl; out[2] = dv; out[3] = ts; out[4] = nb;
}

// ---------------------------------------------------------------------------
extern "C" void kernel_launch(void* const* d_in, const int* in_sizes, int n_in,
                              void* d_out, int out_size, void* d_ws, size_t ws_size,
                              hipStream_t stream) {
    (void)in_sizes; (void)n_in; (void)out_size; (void)ws_size;
    const float* z     = (const float*)d_in[0];
    const float* nodes = (const float*)d_in[1];

    float* out      = (float*)d_out;
    float* som_z    = out + OUT_SOMZ;
    float* scal_out = out + OUT_SCAL;
    float* q        = out + OUT_Q;
    float* bmu_f    = out + OUT_BMU;

    float* wsf      = (float*)d_ws;
    float* colsum   = wsf;
    float* nodenorm = wsf + 1024;
    float* scal     = wsf + 2048;     // kl, div, ts, nb accumulators
    int*   bmu_i    = (int*)(wsf + 2052);

    som_init   <<<4,    256, 0, stream>>>(nodes, colsum, nodenorm, scal);
    som_phase1 <<<256,  256, 0, stream>>>(z, nodes, nodenorm, colsum,
                                          som_z, q, bmu_f, bmu_i);
    som_kl     <<<2048, 256, 0, stream>>>(q, colsum, &scal[0]);
    som_tsmooth<<<2048, 256, 0, stream>>>(z, &scal[2]);
    som_div    <<<64,   256, 0, stream>>>(nodes, nodenorm, &scal[1]);
    som_neigh  <<<128,  256, 0, stream>>>(bmu_i, &scal[3]);
    som_final  <<<1,    1,   0, stream>>>(scal, scal_out);
}